// MemcapacitiveReservoir_10136122819231
// MI455X (gfx1250) — compile-verified
//
#include <hip/hip_runtime.h>
#include <hip/hip_bf16.h>

// ---------------------------------------------------------------------------
// Memcapacitive reservoir, MI455X (gfx1250).
//
// Part 1: drive = x @ Win^T  via V_WMMA_F32_16X16X4_F32 (fp32 matrix cores),
//         written into d_out (it is consumed element-by-element and then
//         overwritten by the recurrence, so no scratch needed for it).
// Part 2: persistent grid-wide kernel; whole W (16 MB) cached in LDS
//         (128 blocks x 128 KB slabs, CDNA5 has 320 KB LDS / WGP), one
//         agent-scope atomic barrier per timestep, phi kept in registers.
// ---------------------------------------------------------------------------

#define R_N      2048
#define R_T      2048
#define R_D      128
#define R_C0     1.0f
#define R_K      0.5f
#define R_DT     1e-3f

#define GBLK     128          // persistent blocks (grid barrier members)
#define BT       256          // threads per block (8 waves of 32)
#define CPB      16           // columns owned per block  (GBLK*CPB == R_N)
#define SLICES   16           // j-dimension slices       (CPB*SLICES == BT)

typedef float v2f __attribute__((ext_vector_type(2)));
typedef float v8f __attribute__((ext_vector_type(8)));

// ---------------------------------------------------------------------------
// Kernel 0: zero the barrier counter and the q ping-pong buffers.
// ---------------------------------------------------------------------------
__global__ void init_ws_kernel(float* q0, float* q1, unsigned int* cnt) {
    int i = blockIdx.x * blockDim.x + threadIdx.x;
    if (i == 0) *cnt = 0u;
    if (i < R_N) { q0[i] = 0.0f; q1[i] = 0.0f; }
}

// ---------------------------------------------------------------------------
// Kernel 1: drive[t][n] = sum_d x[t][d] * Win[n][d]   (fp32 WMMA)
// One wave computes one 16x16 tile of drive. K advances 4 per WMMA.
// A (16x4 f32) layout: lanes 0-15 -> M, v0 holds K=0 (lo)/K=2 (hi), v1 K=1/K=3.
// B (4x16 f32) mirrors it with N across lanes. C/D: 8 VGPRs, M = r + 8*hi.
// ---------------------------------------------------------------------------
__global__ void __launch_bounds__(BT) drive_wmma_kernel(
        const float* __restrict__ x,      // [T, D]
        const float* __restrict__ Win,    // [N, D]
        float* __restrict__ drive) {      // [T, N] (== d_out)
    const int lane = threadIdx.x & 31;
    const int wave = threadIdx.x >> 5;
    const int tile = blockIdx.x * (BT >> 5) + wave;   // 0 .. (128*128-1)
    const int tm   = tile >> 7;                       // tile row over T
    const int tn   = tile & 127;                      // tile col over N
    const int m    = lane & 15;
    const int hi   = lane >> 4;                       // 0 or 1

    const float* arow = x   + (tm * 16 + m) * R_D;    // A row (M = m)
    const float* brow = Win + (tn * 16 + m) * R_D;    // B col (N = m), Win row-major

    v8f c = {};
    #pragma unroll 4
    for (int k0 = 0; k0 < R_D; k0 += 4) {
        v2f a, b;
        a.x = arow[k0 + 2 * hi];
        a.y = arow[k0 + 2 * hi + 1];
        b.x = brow[k0 + 2 * hi];
        b.y = brow[k0 + 2 * hi + 1];
        // (neg_a, A, neg_b, B, c_mod, C, reuse_a, reuse_b)
        c = __builtin_amdgcn_wmma_f32_16x16x4_f32(false, a, false, b,
                                                  (short)0, c, false, false);
    }

    #pragma unroll
    for (int r = 0; r < 8; ++r) {
        drive[(tm * 16 + r + 8 * hi) * R_N + tn * 16 + m] = c[r];
    }
}

// ---------------------------------------------------------------------------
// Kernel 2: persistent recurrence.
//   v      = tanh(q @ W + drive[t])
//   phi   += v * DT
//   q_new  = (C0 + K*phi) * v            -> states[t] (overwrites drive[t])
// W columns for this block live in LDS for the whole kernel.
// ---------------------------------------------------------------------------
__global__ void __launch_bounds__(BT, 1) reservoir_kernel(
        const float* __restrict__ W,      // [N, N] row-major: W[j*N + n]
        float* __restrict__ states,       // [T, N]; holds drive on entry
        float* __restrict__ q0,
        float* __restrict__ q1,
        unsigned int* __restrict__ cnt) {

    __shared__ float wslab[R_N * CPB];    // 128 KB: W[:, block cols]
    __shared__ float qs[R_N];             //   8 KB: staged state vector
    __shared__ float part[BT];            //   1 KB: reduction partials

    const int tx  = threadIdx.x;
    const int c   = tx & (CPB - 1);       // local column 0..15
    const int s   = tx >> 4;              // j-slice     0..15
    const int col = blockIdx.x * CPB + c; // global column

    // One-time W slab load: wslab[j*CPB + c] = W[j][col].
    // For fixed j, lanes with consecutive c read 64B contiguous global memory.
    for (int j = s; j < R_N; j += SLICES)
        wslab[j * CPB + c] = W[j * R_N + col];
    __syncthreads();

    float phi = 0.0f;
    float* qb[2] = { q0, q1 };

    for (int t = 0; t < R_T; ++t) {
        // Stage current q (written device-wide last step) into LDS.
        const float* qcur = qb[t & 1];
        for (int i = tx; i < R_N; i += BT) qs[i] = qcur[i];
        __syncthreads();

        // Partial dot over interleaved slice: j = s + 16*i.
        // 32 lanes of a wave hit 32 distinct LDS banks (stride-16 layout).
        float acc = 0.0f;
        #pragma unroll 8
        for (int i = 0; i < R_N / SLICES; ++i) {
            const int j = s + i * SLICES;
            acc += qs[j] * wslab[j * CPB + c];
        }
        part[tx] = acc;
        __syncthreads();

        // Column owner (slice 0) finishes reduction + state update.
        if (s == 0) {
            float sum = 0.0f;
            #pragma unroll
            for (int ss = 0; ss < SLICES; ++ss) sum += part[ss * CPB + c];
            const float u  = states[t * R_N + col];   // drive (read-then-overwrite)
            const float v  = tanhf(sum + u);
            phi += v * R_DT;
            const float qn = (R_C0 + R_K * phi) * v;
            states[t * R_N + col]   = qn;             // output state
            qb[(t + 1) & 1][col]    = qn;             // next-step input
            __threadfence();                          // publish to agent scope
        }
        __syncthreads();

        // Grid-wide barrier: monotonic counter, one arrival per block.
        if (tx == 0) {
            __hip_atomic_fetch_add(cnt, 1u, __ATOMIC_RELEASE,
                                   __HIP_MEMORY_SCOPE_AGENT);
            const unsigned int target = (unsigned int)(t + 1) * GBLK;
            while (__hip_atomic_load(cnt, __ATOMIC_ACQUIRE,
                                     __HIP_MEMORY_SCOPE_AGENT) < target) {
                __builtin_amdgcn_s_sleep(1);
            }
        }
        __syncthreads();
    }
}

// ---------------------------------------------------------------------------
extern "C" void kernel_launch(void* const* d_in, const int* in_sizes, int n_in,
                              void* d_out, int out_size, void* d_ws, size_t ws_size,
                              hipStream_t stream) {
    const float* x   = (const float*)d_in[0];   // [1, T, D]
    const float* W   = (const float*)d_in[1];   // [N, N]
    const float* Win = (const float*)d_in[2];   // [N, D]
    float* states    = (float*)d_out;           // [1, T, N]

    unsigned int* cnt = (unsigned int*)d_ws;
    float* q0 = (float*)d_ws + 64;              // 256B past counter
    float* q1 = q0 + R_N;

    // 0) zero counter + q ping-pong buffers (ws is not re-poisoned between replays)
    init_ws_kernel<<<8, 256, 0, stream>>>(q0, q1, cnt);

    // 1) drive = x @ Win^T via fp32 WMMA; 16384 tiles, 8 waves/block
    drive_wmma_kernel<<<(R_T / 16) * (R_N / 16) / (BT / 32), BT, 0, stream>>>(
        x, Win, states);

    // 2) persistent recurrence: 128 blocks, whole W resident in LDS
    reservoir_kernel<<<GBLK, BT, 0, stream>>>(W, states, q0, q1, cnt);
}